// ROIAlign_88390426952235
// MI455X (gfx1250) — compile-verified
//
#include <hip/hip_runtime.h>

typedef __attribute__((ext_vector_type(2))) float v2f;
typedef __attribute__((ext_vector_type(8))) float v8f;

#define POOLN 7
#define CCH   256
#define NBOX  1000

// Matches reference sampling exactly:
//  s  = (a + t*(b-a)) * (H-1), t = idx/6
//  i0 = clip((int)floor(s), 0, H-1); i1 = clip(i0+1, 0, H-1); fr = s - floor(s)
__device__ __forceinline__ void sample_axis(float a, float b, int idx, int Hm1,
                                            int& i0, int& i1, float& fr) {
    float t  = (float)idx * (1.0f / 6.0f);
    float s  = (a + t * (b - a)) * (float)Hm1;
    float s0 = floorf(s);
    fr = s - s0;
    int q = (int)s0;
    q  = q < 0 ? 0 : (q > Hm1 ? Hm1 : q);
    i0 = q;
    i1 = (q + 1 > Hm1) ? Hm1 : q + 1;
}

__global__ __launch_bounds__(256) void roialign_wmma_kernel(
    const float* __restrict__ p3, const float* __restrict__ p4,
    const float* __restrict__ p5, const float* __restrict__ p6,
    const float* __restrict__ prop, float* __restrict__ out) {

    const int roi = blockIdx.x;            // 0 .. B*N-1 (2000)
    const int bb  = roi / NBOX;

    // ---- proposal + FPN level selection (round-half-even like jnp.round) ----
    const float by1 = prop[roi * 4 + 0];
    const float bx1 = prop[roi * 4 + 1];
    const float by2 = prop[roi * 4 + 2];
    const float bx2 = prop[roi * 4 + 3];

    float area = (by2 - by1) * (bx2 - bx1);
    area = area > 0.0f ? area : 0.0f;
    // log2( sqrt(area) / (224/1024) + 1e-8 )
    float rl  = log2f(sqrtf(area) * (1024.0f / 224.0f) + 1e-8f);
    int   lvl = 4 + (int)rintf(rl);
    lvl = lvl < 3 ? 3 : (lvl > 6 ? 6 : lvl);

    const float* fmap;
    int H;
    switch (lvl) {
        case 3:  fmap = p3; H = 128; break;
        case 4:  fmap = p4; H = 64;  break;
        case 5:  fmap = p5; H = 32;  break;
        default: fmap = p6; H = 16;  break;
    }
    const float* fb = fmap + (size_t)bb * H * H * CCH;
    const int Hm1 = H - 1;

    const int tid  = threadIdx.x;
    const int lane = tid & 31;
    const int wave = tid >> 5;
    const int m    = lane & 15;   // M row (A/C/D) or N column (B)
    const int hi   = lane >> 4;   // lane half -> K pair selector

    // ---- A matrix: x-interp weights, 16x16 padded, split into 4 K=4 chunks ----
    // A[px, 2*px] = 1-fx[px], A[px, 2*px+1] = fx[px]; rows m >= 7 are zero.
    // Wave32 16x4 f32 A layout: lanes0-15 VGPR0->K0,VGPR1->K1; lanes16-31 -> K2,K3.
    float fxm = 0.0f;
    if (m < POOLN) { int t0, t1; sample_axis(bx1, bx2, m, Hm1, t0, t1, fxm); }

    v2f Amat[4];
    #pragma unroll
    for (int g = 0; g < 4; ++g) {
        const int k0 = 4 * g + 2 * hi;
        float a0 = 0.0f, a1 = 0.0f;
        if (m < POOLN) {
            a0 = (k0     == 2 * m) ? (1.0f - fxm) : ((k0     == 2 * m + 1) ? fxm : 0.0f);
            a1 = (k0 + 1 == 2 * m) ? (1.0f - fxm) : ((k0 + 1 == 2 * m + 1) ? fxm : 0.0f);
        }
        Amat[g].x = a0;
        Amat[g].y = a1;
    }

    // ---- per-lane source columns for B gathers ----
    // B layout (4x16 f32): lanes0-15 VGPR0->K0,VGPR1->K1; lanes16-31 VGPR0->K2,VGPR1->K3.
    // k = 4g + 2*hi + v ; col[k] = (k odd) ? x1[k>>1] : x0[k>>1] ; padded k -> col 0 (A=0).
    int colv[8];
    #pragma unroll
    for (int g = 0; g < 4; ++g) {
        #pragma unroll
        for (int v = 0; v < 2; ++v) {
            const int k   = 4 * g + 2 * hi + v;
            const int pxk = k >> 1;
            int c = 0;
            if (pxk < POOLN) {
                int cx0, cx1; float cf;
                sample_axis(bx1, bx2, pxk, Hm1, cx0, cx1, cf);
                c = (k & 1) ? cx1 : cx0;
            }
            colv[2 * g + v] = c;
        }
    }

    const size_t outbase = (size_t)roi * (POOLN * POOLN * CCH);

    // 16 channel tiles of 16, distributed over 8 waves (2 tiles per wave)
    for (int ct = wave; ct < CCH / 16; ct += 8) {
        const int ch = ct * 16 + m;   // this lane's channel (N index / store column)

        for (int py = 0; py < POOLN; ++py) {
            int ry0, ry1; float fy;
            sample_axis(by1, by2, py, Hm1, ry0, ry1, fy);

            const float* rowT = fb + (size_t)ry0 * H * CCH + ch;
            const float* rowB = fb + (size_t)ry1 * H * CCH + ch;

            v8f accT = {0.f, 0.f, 0.f, 0.f, 0.f, 0.f, 0.f, 0.f};
            v8f accB = {0.f, 0.f, 0.f, 0.f, 0.f, 0.f, 0.f, 0.f};

            #pragma unroll
            for (int g = 0; g < 4; ++g) {
                v2f bT, bB;
                bT.x = rowT[(size_t)colv[2 * g]     * CCH];
                bT.y = rowT[(size_t)colv[2 * g + 1] * CCH];
                bB.x = rowB[(size_t)colv[2 * g]     * CCH];
                bB.y = rowB[(size_t)colv[2 * g + 1] * CCH];
                accT = __builtin_amdgcn_wmma_f32_16x16x4_f32(
                    false, Amat[g], false, bT, (short)0, accT, false, false);
                accB = __builtin_amdgcn_wmma_f32_16x16x4_f32(
                    false, Amat[g], false, bB, (short)0, accB, false, false);
            }

            // y-lerp + store: D layout -> VGPR j holds M=j (px=j) for lanes 0-15
            if (hi == 0) {
                float* o = out + outbase + (size_t)py * (POOLN * CCH) + ch;
                #pragma unroll
                for (int j = 0; j < POOLN; ++j) {
                    const float v = accT[j] + fy * (accB[j] - accT[j]);
                    o[(size_t)j * CCH] = v;
                }
            }
        }
    }
}

extern "C" void kernel_launch(void* const* d_in, const int* in_sizes, int n_in,
                              void* d_out, int out_size, void* d_ws, size_t ws_size,
                              hipStream_t stream) {
    (void)in_sizes; (void)n_in; (void)out_size; (void)d_ws; (void)ws_size;
    const float* p3   = (const float*)d_in[0];
    const float* p4   = (const float*)d_in[1];
    const float* p5   = (const float*)d_in[2];
    const float* p6   = (const float*)d_in[3];
    const float* prop = (const float*)d_in[4];
    float* outp = (float*)d_out;

    roialign_wmma_kernel<<<dim3(2 * NBOX), dim3(256), 0, stream>>>(
        p3, p4, p5, p6, prop, outp);
}